// IdeficsPerceiverResampler_13451837571594
// MI455X (gfx1250) — compile-verified
//
#include <hip/hip_runtime.h>
#include <math.h>

#define DEPTH 6
#define E     1280
#define H     16
#define HD    96
#define INNER 1536
#define NL    64
#define IM    5120
#define BB    32
#define SS    576
#define TT    (SS + NL)      // 640
#define EPSF  1e-5f
#define QSCALE 0.10206207261596575f   // 96^-0.5

typedef float v2f __attribute__((ext_vector_type(2)));
typedef float v8f __attribute__((ext_vector_type(8)));
typedef unsigned int u32;
typedef unsigned long long u64;
typedef u32 v4u __attribute__((ext_vector_type(4)));
typedef int v4i __attribute__((ext_vector_type(4)));
typedef int v8i __attribute__((ext_vector_type(8)));

__device__ __forceinline__ v8f wmma4(v2f a, v2f b, v8f c) {
    // D = A(16x4 f32) * B(4x16 f32) + C(16x16 f32)
    return __builtin_amdgcn_wmma_f32_16x16x4_f32(
        false, a, false, b, (short)0, c, false, false);
}

// TDM group1 word0 flags: data_size=4B (2<<16); A additionally pads 1 DWORD
// every 32 DWORDs (pad_enable<<20 | pad_interval=4<<22 | pad_amount=0<<25)
// -> reproduces the As[...][33] anti-bank-conflict stride in hardware.
#define TDM_FLAGS_A ((2 << 16) | (1 << 20) | (4 << 22))
#define TDM_FLAGS_B (2 << 16)

// Issue one 2D-tile TENSOR_LOAD_TO_LDS: tile_x elems (contiguous) x tile_y
// rows, row stride `stride_elems`, fp32 elements. One issue per wave.
// (amdgpu-toolchain clang-23: 6-arg builtin, extra int32x8 group before cpol)
__device__ __forceinline__ void tdm_load_tile(
    u32 lds_byte_off, const void* gsrc, int tile_x, int tile_y,
    u64 stride_elems, int flags)
{
    u64 ga = (u64)(uintptr_t)gsrc;
    v4u g0;
    g0[0] = 1u;                                   // count=1, user descriptor
    g0[1] = lds_byte_off;                         // LDS dest (bytes)
    g0[2] = (u32)ga;                              // global_addr[31:0]
    g0[3] = (u32)((ga >> 32) & 0x01FFFFFFull)     // global_addr[56:32]
          | (2u << 30);                           // type = 2 ("image")
    v8i g1;
    g1[0] = flags;                                // mask=0, data_size, pad cfg
    g1[1] = 0;                                    // abar=0, tensor_dim0 lo16(=0)
    g1[2] = 0x4000;                               // tensor_dim0=0x40000000, dim1 lo16
    g1[3] = 0x4000 | (tile_x << 16);              // tensor_dim1 hi16, tile_dim0
    g1[4] = tile_y & 0xFFFF;                      // tile_dim1, tile_dim2=0
    g1[5] = (int)(u32)(stride_elems & 0xFFFFFFFFull);   // dim0_stride[31:0]
    g1[6] = (int)(u32)((stride_elems >> 32) & 0xFFFFull); // [47:32], dim1_stride=0
    g1[7] = 0;
    v4i z4 = {0, 0, 0, 0};
    v8i z8 = {0, 0, 0, 0, 0, 0, 0, 0};
    __builtin_amdgcn_tensor_load_to_lds(g0, g1, z4, z4, z8, 0);
}

// ---------------------------------------------------------------------------
// Row LayerNorm over width E=1280. block=256, one row per block.
// mode 0: out[r] = LN(in[r])
// mode 1: ctx -> kv   (in row r = b*S+s, out row = b*T+s)
// mode 2: lat -> kv & compact (out row = b*T+S+nl, out2 row = r)
// ---------------------------------------------------------------------------
__global__ __launch_bounds__(256) void rowln_kernel(
    const float* __restrict__ x, const float* __restrict__ w,
    const float* __restrict__ bias, float* __restrict__ out,
    float* __restrict__ out2, int mode)
{
    __shared__ float red[256];
    const int r   = blockIdx.x;
    const int tid = threadIdx.x;

    const float* xin = x + (size_t)r * E;
    float xv[5];
    float s = 0.f;
    #pragma unroll
    for (int i = 0; i < 5; ++i) { xv[i] = xin[tid + i * 256]; s += xv[i]; }

    red[tid] = s; __syncthreads();
    for (int o = 128; o > 0; o >>= 1) { if (tid < o) red[tid] += red[tid + o]; __syncthreads(); }
    const float mu = red[0] * (1.0f / (float)E);
    __syncthreads();

    float sq = 0.f;
    #pragma unroll
    for (int i = 0; i < 5; ++i) { float d = xv[i] - mu; sq += d * d; }
    red[tid] = sq; __syncthreads();
    for (int o = 128; o > 0; o >>= 1) { if (tid < o) red[tid] += red[tid + o]; __syncthreads(); }
    const float rstd = rsqrtf(red[0] * (1.0f / (float)E) + EPSF);

    size_t orow, orow2 = 0;
    bool dual = false;
    if (mode == 0) {
        orow = (size_t)r;
    } else if (mode == 1) {
        int b_ = r / SS, s_ = r % SS;
        orow = (size_t)b_ * TT + s_;
    } else {
        int b_ = r / NL, nl = r % NL;
        orow = (size_t)b_ * TT + SS + nl;
        orow2 = (size_t)r;
        dual = true;
    }
    float* op  = out + orow * E;
    float* op2 = dual ? (out2 + orow2 * E) : nullptr;
    #pragma unroll
    for (int i = 0; i < 5; ++i) {
        int c = tid + i * 256;
        float y = (xv[i] - mu) * rstd * w[c] + bias[c];
        op[c] = y;
        if (dual) op2[c] = y;
    }
}

// ---------------------------------------------------------------------------
// Head LayerNorm over contiguous 96-float segments, one wave per segment.
// Folds an extra scale (q: HD^-0.5, k: 1.0). In-place.
// ---------------------------------------------------------------------------
__global__ __launch_bounds__(256) void headln_kernel(
    float* __restrict__ x, const float* __restrict__ w,
    const float* __restrict__ bias, long nseg, float scale)
{
    const int tid = threadIdx.x;
    const long seg = (long)blockIdx.x * 8 + (tid >> 5);
    if (seg >= nseg) return;
    const int l = tid & 31;
    float* p = x + seg * (long)HD;

    float v0 = p[l], v1 = p[l + 32], v2 = p[l + 64];
    float s = v0 + v1 + v2;
    #pragma unroll
    for (int o = 16; o > 0; o >>= 1) s += __shfl_xor(s, o, 32);
    const float mu = s * (1.0f / (float)HD);
    float d0 = v0 - mu, d1 = v1 - mu, d2 = v2 - mu;
    float sq = d0 * d0 + d1 * d1 + d2 * d2;
    #pragma unroll
    for (int o = 16; o > 0; o >>= 1) sq += __shfl_xor(sq, o, 32);
    const float rstd = rsqrtf(sq * (1.0f / (float)HD) + EPSF);

    p[l]      = (d0 * rstd * w[l]      + bias[l])      * scale;
    p[l + 32] = (d1 * rstd * w[l + 32] + bias[l + 32]) * scale;
    p[l + 64] = (d2 * rstd * w[l + 64] + bias[l + 64]) * scale;
}

// ---------------------------------------------------------------------------
// WMMA fp32 GEMM with TDM double-buffered LDS staging.
// C[MxN] = A[MxK] * B[KxN] (+ epilogue). Macro-tile 128(M) x 64(N), 8 waves.
// Wave 0 issues TENSOR_LOAD_TO_LDS for buffer p^1 while all waves run WMMAs
// on buffer p. M%128==0, N%64==0, K%32==0. epi: 0=store, 1=+Res, 2=ReLU.
// ---------------------------------------------------------------------------
__global__ __launch_bounds__(256) void gemm_kernel(
    const float* __restrict__ A, const float* __restrict__ Bw,
    const float* __restrict__ Res, float* __restrict__ C,
    int M, int N, int K, int epi)
{
    __shared__ float As[2][128][33];   // 33-stride produced by TDM pad cfg
    __shared__ float Bs[2][32][64];

    const int tid  = threadIdx.x;
    const int w    = tid >> 5;
    const int l    = tid & 31;
    const int lh   = l & 15;
    const int koff = (l < 16) ? 0 : 2;
    const int m0   = blockIdx.y * 128;
    const int n0   = blockIdx.x * 64;

    const u32 ldsA[2] = { (u32)(uintptr_t)&As[0][0][0], (u32)(uintptr_t)&As[1][0][0] };
    const u32 ldsB[2] = { (u32)(uintptr_t)&Bs[0][0][0], (u32)(uintptr_t)&Bs[1][0][0] };

    v8f acc[4];
    #pragma unroll
    for (int nt = 0; nt < 4; ++nt)
        #pragma unroll
        for (int j = 0; j < 8; ++j) acc[nt][j] = 0.f;

    // prologue: DMA first K-slab into buffer 0
    if (tid < 32) {
        tdm_load_tile(ldsA[0], A + (size_t)m0 * K, 32, 128, (u64)K, TDM_FLAGS_A);
        tdm_load_tile(ldsB[0], Bw + n0,            64, 32,  (u64)N, TDM_FLAGS_B);
    }

    int buf = 0;
    for (int k0 = 0; k0 < K; k0 += 32) {
        if (tid < 32) __builtin_amdgcn_s_wait_tensorcnt(0);
        __syncthreads();   // buffer `buf` ready for all; prev compute done
        if (tid < 32 && (k0 + 32) < K) {
            tdm_load_tile(ldsA[buf ^ 1], A + (size_t)m0 * K + k0 + 32,
                          32, 128, (u64)K, TDM_FLAGS_A);
            tdm_load_tile(ldsB[buf ^ 1], Bw + (size_t)(k0 + 32) * N + n0,
                          64, 32, (u64)N, TDM_FLAGS_B);
        }
        #pragma unroll
        for (int s = 0; s < 8; ++s) {
            v2f a;
            a.x = As[buf][w * 16 + lh][s * 4 + koff];
            a.y = As[buf][w * 16 + lh][s * 4 + koff + 1];
            #pragma unroll
            for (int nt = 0; nt < 4; ++nt) {
                v2f b;
                b.x = Bs[buf][s * 4 + koff][nt * 16 + lh];
                b.y = Bs[buf][s * 4 + koff + 1][nt * 16 + lh];
                acc[nt] = wmma4(a, b, acc[nt]);
            }
        }
        __syncthreads();   // done reading `buf` before TDM refills it next iter
        buf ^= 1;
    }

    #pragma unroll
    for (int nt = 0; nt < 4; ++nt) {
        #pragma unroll
        for (int j = 0; j < 8; ++j) {
            int row = m0 + w * 16 + ((l < 16) ? j : j + 8);
            int col = n0 + nt * 16 + lh;
            float v = acc[nt][j];
            if (epi == 1)      v += Res[(size_t)row * N + col];
            else if (epi == 2) v = fmaxf(v, 0.f);
            C[(size_t)row * N + col] = v;
        }
    }
}

// ---------------------------------------------------------------------------
// Flash-style perceiver attention, one block per (b,h).
// q[B*NL, INNER] (LN'd, pre-scaled), k/v[B*T, INNER] (k LN'd).
// o[B*NL, INNER]  (o[:, h*HD .. ]).
// ---------------------------------------------------------------------------
__global__ __launch_bounds__(256) void attn_kernel(
    const float* __restrict__ q, const float* __restrict__ k,
    const float* __restrict__ v, float* __restrict__ o)
{
    __shared__ float kv[64][97];      // k tile then v tile  [key][hd]
    __shared__ float Sm[64][65];      // scores / probs      [m][key]
    __shared__ float rscale[64];
    __shared__ float linv[64];

    const int b   = blockIdx.x / H;
    const int h   = blockIdx.x % H;
    const int tid = threadIdx.x;
    const int w   = tid >> 5;
    const int l   = tid & 31;
    const int lh  = l & 15;
    const int koff = (l < 16) ? 0 : 2;
    const int mt  = w >> 1;                 // M-tile (0..3), latent rows mt*16..+15
    const int mrow = mt * 16 + lh;          // A-fragment row for this lane

    // q A-fragments in registers (24 k-steps over HD=96)
    v2f qa[24];
    {
        const float* qp = q + ((size_t)b * NL + mrow) * INNER + h * HD;
        #pragma unroll
        for (int s = 0; s < 24; ++s) { qa[s].x = qp[s * 4 + koff]; qa[s].y = qp[s * 4 + koff + 1]; }
    }

    v8f oa[3];
    #pragma unroll
    for (int c = 0; c < 3; ++c)
        #pragma unroll
        for (int j = 0; j < 8; ++j) oa[c][j] = 0.f;

    float mprev = -3.0e38f, lrun = 0.f;     // live in threads 0..63

    for (int t = 0; t < 10; ++t) {
        // ---- stage k tile ----
        for (int idx = tid; idx < 64 * 96; idx += 256) {
            int r = idx / 96, c = idx % 96;
            kv[r][c] = k[((size_t)b * TT + t * 64 + r) * INNER + h * HD + c];
        }
        __syncthreads();

        // ---- GEMM1: S = q * k^T (per wave: 2 of 16 tiles) ----
        v8f sacc[2];
        #pragma unroll
        for (int nt = 0; nt < 2; ++nt)
            #pragma unroll
            for (int j = 0; j < 8; ++j) sacc[nt][j] = 0.f;
        #pragma unroll
        for (int s = 0; s < 24; ++s) {
            v2f a = qa[s];
            #pragma unroll
            for (int nt = 0; nt < 2; ++nt) {
                int n = ((w & 1) * 2 + nt) * 16 + lh;
                v2f bb;
                bb.x = kv[n][s * 4 + koff];
                bb.y = kv[n][s * 4 + koff + 1];
                sacc[nt] = wmma4(a, bb, sacc[nt]);
            }
        }
        #pragma unroll
        for (int nt = 0; nt < 2; ++nt)
            #pragma unroll
            for (int j = 0; j < 8; ++j)
                Sm[mt * 16 + ((l < 16) ? j : j + 8)][((w & 1) * 2 + nt) * 16 + lh] = sacc[nt][j];
        __syncthreads();

        // ---- online softmax (one thread per latent row) ----
        if (tid < 64) {
            float mx = mprev;
            #pragma unroll 8
            for (int i = 0; i < 64; ++i) mx = fmaxf(mx, Sm[tid][i]);
            float scl = __expf(mprev - mx);
            float sum = 0.f;
            #pragma unroll 8
            for (int i = 0; i < 64; ++i) {
                float p = __expf(Sm[tid][i] - mx);
                Sm[tid][i] = p;
                sum += p;
            }
            lrun = lrun * scl + sum;
            mprev = mx;
            rscale[tid] = scl;
        }
        __syncthreads();

        // ---- rescale running o ----
        #pragma unroll
        for (int j = 0; j < 8; ++j) {
            float sj = rscale[mt * 16 + ((l < 16) ? j : j + 8)];
            #pragma unroll
            for (int c = 0; c < 3; ++c) oa[c][j] *= sj;
        }

        // ---- stage v tile (reuse kv) ----
        for (int idx = tid; idx < 64 * 96; idx += 256) {
            int r = idx / 96, c = idx % 96;
            kv[r][c] = v[((size_t)b * TT + t * 64 + r) * INNER + h * HD + c];
        }
        __syncthreads();

        // ---- GEMM2: o += P * v (per wave: 3 of 24 tiles, K=64) ----
        #pragma unroll
        for (int s = 0; s < 16; ++s) {
            v2f a;
            a.x = Sm[mrow][s * 4 + koff];
            a.y = Sm[mrow][s * 4 + koff + 1];
            #pragma unroll
            for (int c = 0; c < 3; ++c) {
                int n = ((w & 1) * 3 + c) * 16 + lh;
                v2f bb;
                bb.x = kv[s * 4 + koff][n];
                bb.y = kv[s * 4 + koff + 1][n];
                oa[c] = wmma4(a, bb, oa[c]);
            }
        }
        __syncthreads();
    }

    if (tid < 64) linv[tid] = 1.0f / lrun;
    __syncthreads();

    #pragma unroll
    for (int j = 0; j < 8; ++j) {
        int rl = mt * 16 + ((l < 16) ? j : j + 8);
        float inv = linv[rl];
        #pragma unroll
        for (int c = 0; c < 3; ++c)
            o[((size_t)b * NL + rl) * INNER + h * HD + ((w & 1) * 3 + c) * 16 + lh] = oa[c][j] * inv;
    }
}

// ---------------------------------------------------------------------------
// lat[b,nl,:] = latents[nl,:]
// ---------------------------------------------------------------------------
__global__ __launch_bounds__(256) void latinit_kernel(
    const float* __restrict__ latents, float* __restrict__ lat)
{
    size_t idx = (size_t)blockIdx.x * 256 + threadIdx.x;
    if (idx < (size_t)BB * NL * E) lat[idx] = latents[idx % ((size_t)NL * E)];
}

// ---------------------------------------------------------------------------
extern "C" void kernel_launch(void* const* d_in, const int* in_sizes, int n_in,
                              void* d_out, int out_size, void* d_ws, size_t ws_size,
                              hipStream_t stream) {
    const float* context  = (const float*)d_in[0];
    const float* latents  = (const float*)d_in[1];
    const float* ctx_ln_w = (const float*)d_in[2];
    const float* ctx_ln_b = (const float*)d_in[3];
    const float* lat_ln_w = (const float*)d_in[4];
    const float* lat_ln_b = (const float*)d_in[5];
    const float* q_ln_w   = (const float*)d_in[6];
    const float* q_ln_b   = (const float*)d_in[7];
    const float* k_ln_w   = (const float*)d_in[8];
    const float* k_ln_b   = (const float*)d_in[9];
    const float* Wq       = (const float*)d_in[10];
    const float* Wk       = (const float*)d_in[11];
    const float* Wv       = (const float*)d_in[12];
    const float* Wo       = (const float*)d_in[13];
    const float* mlp_ln_w = (const float*)d_in[14];
    const float* mlp_ln_b = (const float*)d_in[15];
    const float* Wfc      = (const float*)d_in[16];
    const float* Wproj    = (const float*)d_in[17];
    const float* out_ln_w = (const float*)d_in[18];
    const float* out_ln_b = (const float*)d_in[19];
    float* out = (float*)d_out;

    // workspace carve-up (floats)
    float* ws   = (float*)d_ws;
    float* lat  = ws;                       // [B*NL, E]
    float* lln  = lat + (size_t)BB*NL*E;    // [B*NL, E]
    float* kvb  = lln + (size_t)BB*NL*E;    // [B*T,  E]
    float* qb   = kvb + (size_t)BB*TT*E;    // [B*NL, INNER]
    float* kb   = qb  + (size_t)BB*NL*INNER;// [B*T,  INNER]
    float* vb   = kb  + (size_t)BB*TT*INNER;// [B*T,  INNER]
    float* ob   = vb  + (size_t)BB*TT*INNER;// [B*NL, INNER]
    float* hb   = ob  + (size_t)BB*NL*INNER;// [B*NL, IM]
    (void)ws_size; (void)in_sizes; (void)n_in; (void)out_size;

    // lat = broadcast(latents)
    latinit_kernel<<<(BB * NL * E + 255) / 256, 256, 0, stream>>>(latents, lat);

    for (int i = 0; i < DEPTH; ++i) {
        const float* Wq_i = Wq + (size_t)i * E * INNER;
        const float* Wk_i = Wk + (size_t)i * E * INNER;
        const float* Wv_i = Wv + (size_t)i * E * INNER;
        const float* Wo_i = Wo + (size_t)i * INNER * E;
        const float* Wf_i = Wfc + (size_t)i * E * IM;
        const float* Wp_i = Wproj + (size_t)i * IM * E;

        // LNs -> kv_in concat + compact latent-LN
        rowln_kernel<<<BB * SS, 256, 0, stream>>>(context, ctx_ln_w + i * E, ctx_ln_b + i * E,
                                                  kvb, nullptr, 1);
        rowln_kernel<<<BB * NL, 256, 0, stream>>>(lat, lat_ln_w + i * E, lat_ln_b + i * E,
                                                  kvb, lln, 2);
        // projections
        gemm_kernel<<<dim3(INNER / 64, (BB * NL) / 128), 256, 0, stream>>>(
            lln, Wq_i, nullptr, qb, BB * NL, INNER, E, 0);
        gemm_kernel<<<dim3(INNER / 64, (BB * TT) / 128), 256, 0, stream>>>(
            kvb, Wk_i, nullptr, kb, BB * TT, INNER, E, 0);
        gemm_kernel<<<dim3(INNER / 64, (BB * TT) / 128), 256, 0, stream>>>(
            kvb, Wv_i, nullptr, vb, BB * TT, INNER, E, 0);
        // per-head LN (q gets the attention scale folded in)
        headln_kernel<<<(BB * NL * H) / 8, 256, 0, stream>>>(
            qb, q_ln_w + i * HD, q_ln_b + i * HD, (long)BB * NL * H, QSCALE);
        headln_kernel<<<(BB * TT * H) / 8, 256, 0, stream>>>(
            kb, k_ln_w + i * HD, k_ln_b + i * HD, (long)BB * TT * H, 1.0f);
        // attention
        attn_kernel<<<BB * H, 256, 0, stream>>>(qb, kb, vb, ob);
        // output proj + residual -> lat
        gemm_kernel<<<dim3(E / 64, (BB * NL) / 128), 256, 0, stream>>>(
            ob, Wo_i, lat, lat, BB * NL, E, INNER, 1);
        // MLP
        rowln_kernel<<<BB * NL, 256, 0, stream>>>(lat, mlp_ln_w + i * E, mlp_ln_b + i * E,
                                                  lln, nullptr, 0);
        gemm_kernel<<<dim3(IM / 64, (BB * NL) / 128), 256, 0, stream>>>(
            lln, Wf_i, nullptr, hb, BB * NL, IM, E, 2);
        gemm_kernel<<<dim3(E / 64, (BB * NL) / 128), 256, 0, stream>>>(
            hb, Wp_i, lat, lat, BB * NL, E, IM, 1);
    }
    // final LN -> out
    rowln_kernel<<<BB * NL, 256, 0, stream>>>(lat, out_ln_w, out_ln_b, out, nullptr, 0);
}